// SGNet_56831007261231
// MI455X (gfx1250) — compile-verified
//
#include <hip/hip_runtime.h>
#include <hip/hip_bf16.h>
#include <math.h>

#define NN 50000
#define EE 600000
#define DD 128
#define CC 32

typedef float v2f __attribute__((ext_vector_type(2)));
typedef float v8f __attribute__((ext_vector_type(8)));

// ---------------- degree / norm precompute ----------------

__global__ void k_init_deg(float* __restrict__ deg) {
    int i = blockIdx.x * blockDim.x + threadIdx.x;
    if (i < NN) deg[i] = 1.0f;  // self-loop
}

__global__ void k_accum_deg(const int* __restrict__ dst, float* __restrict__ deg) {
    int e = blockIdx.x * blockDim.x + threadIdx.x;
    if (e < EE) unsafeAtomicAdd(&deg[dst[e]], 1.0f);
}

__global__ void k_dinv(float* __restrict__ deg) {
    int i = blockIdx.x * blockDim.x + threadIdx.x;
    if (i < NN) deg[i] = rsqrtf(deg[i]);
}

__global__ void k_enorm(const int* __restrict__ src, const int* __restrict__ dst,
                        const float* __restrict__ dinv, float* __restrict__ enorm) {
    int e = blockIdx.x * blockDim.x + threadIdx.x;
    if (e < EE) enorm[e] = dinv[src[e]] * dinv[dst[e]];
}

// ---------------- propagation: out = in*snorm + scatter_add(in[src]*enorm, dst) ----

// Initialize out[i,:] = in[i,:] * dinv[i]^2   (self-loop term); one float4 per thread.
__global__ void k_prop_init(const float* __restrict__ X, const float* __restrict__ dinv,
                            float* __restrict__ out) {
    int t = blockIdx.x * blockDim.x + threadIdx.x;
    int node = t >> 5, lane = t & 31;
    if (node >= NN) return;
    float s = dinv[node];
    s *= s;
    float4 v = *((const float4*)(X + (size_t)node * DD) + lane);
    v.x *= s; v.y *= s; v.z *= s; v.w *= s;
    *((float4*)(out + (size_t)node * DD) + lane) = v;
}

// One wave per edge: 32 lanes x float4 = 128 floats. Coalesced 512B gather from
// x[src], then 4 native f32 atomic adds per lane into out[dst].
__global__ void k_prop_edges(const float* __restrict__ X, const int* __restrict__ src,
                             const int* __restrict__ dst, const float* __restrict__ enorm,
                             float* __restrict__ out) {
    int t = blockIdx.x * blockDim.x + threadIdx.x;
    int e = t >> 5, lane = t & 31;
    if (e >= EE) return;
    int s = src[e], d = dst[e];
    float en = enorm[e];
    float4 v = *((const float4*)(X + (size_t)s * DD) + lane);
    float* op = out + (size_t)d * DD + lane * 4;
    unsafeAtomicAdd(op + 0, v.x * en);
    unsafeAtomicAdd(op + 1, v.y * en);
    unsafeAtomicAdd(op + 2, v.z * en);
    unsafeAtomicAdd(op + 3, v.w * en);
}

// ---------------- FP32 WMMA GEMM: out = act(X[NN x 128] @ W[128 x nCols] + bias) ----
// ACT: 0 = ELU, 1 = ReLU. One wave per 16x16 output tile; K-loop of 16x16x4 WMMAs.
// f32 A 16x4 layout:  lane<16 -> row lm, K = {k0, k0+1}; lane>=16 -> K = {k0+2, k0+3}
// f32 B 4x16 layout:  mirrored (lane = column, VGPR/half = K)
// f32 C/D layout:     VGPR v holds M = v + 8*(lane>=16), N = lane%16
template <int ACT>
__global__ void k_gemm_wmma(const float* __restrict__ X, const float* __restrict__ W,
                            const float* __restrict__ bias, float* __restrict__ out,
                            int nCols) {
    const int lane = threadIdx.x & 31;
    const int wave = (blockIdx.x * blockDim.x + threadIdx.x) >> 5;
    const int colTiles = nCols >> 4;
    const int totalTiles = (NN >> 4) * colTiles;   // 50000 % 16 == 0, no tail
    if (wave >= totalTiles) return;                // wave-uniform: EXEC all-ones for WMMA
    const int row0 = (wave / colTiles) << 4;
    const int col0 = (wave % colTiles) << 4;
    const int half = lane >> 4;
    const int lm = lane & 15;

    v8f acc = {};
    const float* arow = X + (size_t)(row0 + lm) * DD + half * 2;
    const float* bcol = W + (size_t)(half * 2) * nCols + col0 + lm;
#pragma unroll 8
    for (int k0 = 0; k0 < DD; k0 += 4) {
        v2f a, b;
        a.x = arow[k0];
        a.y = arow[k0 + 1];
        b.x = bcol[(size_t)k0 * nCols];
        b.y = bcol[(size_t)(k0 + 1) * nCols];
        acc = __builtin_amdgcn_wmma_f32_16x16x4_f32(false, a, false, b,
                                                    (short)0, acc, false, false);
    }

    float bv = bias[col0 + lm];
    float* orow = out + (size_t)(row0 + half * 8) * nCols + col0 + lm;
#pragma unroll
    for (int v = 0; v < 8; ++v) {
        float val = acc[v] + bv;
        if (ACT == 0) val = val > 0.0f ? val : expm1f(val);   // ELU(alpha=1)
        else          val = fmaxf(val, 0.0f);                 // ReLU
        orow[(size_t)v * nCols] = val;
    }
}

// ---------------- host-side orchestration ----------------

extern "C" void kernel_launch(void* const* d_in, const int* in_sizes, int n_in,
                              void* d_out, int out_size, void* d_ws, size_t ws_size,
                              hipStream_t stream) {
    const float* x   = (const float*)d_in[0];
    const int*   ei  = (const int*)d_in[1];
    const int*   src = ei;
    const int*   dst = ei + EE;
    const float* W1  = (const float*)d_in[2];
    const float* b1  = (const float*)d_in[3];
    const float* W2  = (const float*)d_in[4];
    const float* b2  = (const float*)d_in[5];
    const float* Wm  = (const float*)d_in[6];
    const float* bm  = (const float*)d_in[7];
    float* out = (float*)d_out;

    // workspace layout: two N x D ping-pong buffers, dinv[N], enorm[E]  (~54 MB)
    float* bufA  = (float*)d_ws;
    float* bufB  = bufA + (size_t)NN * DD;
    float* dinv  = bufB + (size_t)NN * DD;
    float* enorm = dinv + NN;

    const int TB = 256;
    k_init_deg <<<(NN + TB - 1) / TB, TB, 0, stream>>>(dinv);
    k_accum_deg<<<(EE + TB - 1) / TB, TB, 0, stream>>>(dst, dinv);
    k_dinv     <<<(NN + TB - 1) / TB, TB, 0, stream>>>(dinv);
    k_enorm    <<<(EE + TB - 1) / TB, TB, 0, stream>>>(src, dst, dinv, enorm);

    const int gInit  = (NN * 32 + TB - 1) / TB;
    const int gEdges = (int)(((long)EE * 32 + TB - 1) / TB);

    auto prop = [&](const float* in, float* o) {
        k_prop_init <<<gInit,  TB, 0, stream>>>(in, dinv, o);
        k_prop_edges<<<gEdges, TB, 0, stream>>>(in, src, dst, enorm, o);
    };

    const int g128 = ((NN / 16) * (DD / 16) + 7) / 8;  // 8 waves per 256-thread block
    const int g32  = ((NN / 16) * (CC / 16) + 7) / 8;

    prop(x, bufA);                                             // hop 1, layer 1
    prop(bufA, bufB);                                          // hop 2, layer 1
    k_gemm_wmma<0><<<g128, TB, 0, stream>>>(bufB, W1, b1, bufA, DD);  // @W1+b1, ELU
    prop(bufA, bufB);                                          // hop 1, layer 2
    prop(bufB, bufA);                                          // hop 2, layer 2
    k_gemm_wmma<0><<<g128, TB, 0, stream>>>(bufA, W2, b2, bufB, DD);  // @W2+b2, ELU
    k_gemm_wmma<1><<<g32,  TB, 0, stream>>>(bufB, Wm, bm, out, CC);   // @Wm+bm, ReLU
}